// StaticEncoderONNX_420906795077
// MI455X (gfx1250) — compile-verified
//
#include <hip/hip_runtime.h>

// CDNA5 / gfx1250: wave32, WMMA f32 16x16x4.
typedef __attribute__((ext_vector_type(2))) float v2f;
typedef __attribute__((ext_vector_type(8))) float v8f;

#define HDIM 192
#define DDIM 10
#define WAVES_PER_BLOCK 4
#define CT_PER_WAVE 3   // 4 waves * 3 col-tiles * 16 cols = 192 = H

__global__ __launch_bounds__(WAVES_PER_BLOCK * 32)
void static_encoder_kernel(const float* __restrict__ x,
                           const float* __restrict__ W,
                           const float* __restrict__ bias,
                           float* __restrict__ out,
                           int M /* = B*P rows */)
{
    const int tid    = threadIdx.x;
    const int wave   = tid >> 5;
    const int lane   = tid & 31;
    const int laneLo = lane & 15;
    const bool hi    = (lane >= 16);
    const int kbase  = hi ? 2 : 0;     // ISA A-layout: lanes 0-15 hold K={0,1}, 16-31 hold K={2,3}

    const long row0 = (long)blockIdx.x * 16;
    const long rowA = row0 + laneLo;

    // ---- A fragments: 16x4 f32, three K-steps (K = 0..3, 4..7, 8..11 with 10,11 zero) ----
    const float* xr = x + rowA * DDIM;
    v2f a0, a1, a2;
    a0[0] = xr[kbase + 0];     a0[1] = xr[kbase + 1];
    a1[0] = xr[4 + kbase + 0]; a1[1] = xr[4 + kbase + 1];
    {   // K=8,9 live in lanes 0-15; K=10,11 are zero padding (lanes 16-31).
        float x8 = xr[8], x9 = xr[9];
        a2[0] = hi ? 0.0f : x8;
        a2[1] = hi ? 0.0f : x9;
    }

    // ---- row all-zero mask from the A fragments (each lane-half covers disjoint K) ----
    unsigned nz = (unsigned)(a0[0] != 0.0f) | (unsigned)(a0[1] != 0.0f) |
                  (unsigned)(a1[0] != 0.0f) | (unsigned)(a1[1] != 0.0f) |
                  (unsigned)(a2[0] != 0.0f) | (unsigned)(a2[1] != 0.0f);
    nz |= (unsigned)__shfl_xor((int)nz, 16, 32);   // now lane l holds nz of row (l & 15)

    // keep[r] = 1.0 if accumulator row (r + hi*8) has any nonzero input, else 0.0
    float nzf = (float)nz;
    float keep[8];
#pragma unroll
    for (int r = 0; r < 8; ++r)
        keep[r] = __shfl(nzf, r + (hi ? 8 : 0), 32);

    // ---- GEMM: 3 column tiles per wave, 3 WMMAs (K-steps) each ----
#pragma unroll
    for (int c = 0; c < CT_PER_WAVE; ++c) {
        const int ct  = wave * CT_PER_WAVE + c;
        const int col = ct * 16 + laneLo;

        // B fragment: B[k][n] = W[n][k]; lanes 0-15 -> K={0,1}, lanes 16-31 -> K={2,3}
        const float* wr = W + (long)col * DDIM;
        v2f b0, b1, b2;
        b0[0] = wr[kbase + 0];     b0[1] = wr[kbase + 1];
        b1[0] = wr[4 + kbase + 0]; b1[1] = wr[4 + kbase + 1];
        {
            float w8 = wr[8], w9 = wr[9];
            b2[0] = hi ? 0.0f : w8;
            b2[1] = hi ? 0.0f : w9;
        }

        // Seed C with the bias: same value for every row of this column.
        const float bv = bias[col];
        v8f acc;
#pragma unroll
        for (int r = 0; r < 8; ++r) acc[r] = bv;

        // D = A*B + C  (v_wmma_f32_16x16x4_f32), EXEC is all-ones here (no divergence).
        acc = __builtin_amdgcn_wmma_f32_16x16x4_f32(false, a0, false, b0,
                                                    (short)0, acc, false, false);
        acc = __builtin_amdgcn_wmma_f32_16x16x4_f32(false, a1, false, b1,
                                                    (short)0, acc, false, false);
        acc = __builtin_amdgcn_wmma_f32_16x16x4_f32(false, a2, false, b2,
                                                    (short)0, acc, false, false);

        // Store: VGPR r, lanes 0-15 -> row r, lanes 16-31 -> row r+8; cols contiguous
        // => 64B coalesced per half-wave. Multiply by keep[] to zero masked rows.
#pragma unroll
        for (int r = 0; r < 8; ++r) {
            const long row = row0 + r + (hi ? 8 : 0);
            out[row * HDIM + col] = acc[r] * keep[r];
        }
    }

    // ---- mask_p (float 0/1) and pos (7 floats/row): one lane per row, wave 0 only ----
    if (wave == 0 && !hi) {
        float* maskp = out + (long)M * HDIM;
        float* posp  = maskp + M;

        maskp[rowA] = nz ? 0.0f : 1.0f;   // mask_p = (row entirely zero)

        const float p0 = xr[0], p1 = xr[1], p2 = xr[2], p3 = xr[3];
        float* pp = posp + rowA * 7;
        pp[0] = p0;  pp[1] = p1;  pp[2] = p2;  pp[3] = p3;
        pp[4] = 0.0f; pp[5] = 1.0f; pp[6] = 0.0f;
    }
}

extern "C" void kernel_launch(void* const* d_in, const int* in_sizes, int n_in,
                              void* d_out, int out_size, void* d_ws, size_t ws_size,
                              hipStream_t stream) {
    const float* x = (const float*)d_in[0];  // (B, P, D) f32
    const float* W = (const float*)d_in[1];  // (H, D)    f32
    const float* b = (const float*)d_in[2];  // (H,)      f32
    float* out = (float*)d_out;              // x_result | mask_p | pos (flat, f32)

    const int M = in_sizes[0] / DDIM;        // B*P = 524288
    const int tiles = M / 16;                // 32768 row tiles

    static_encoder_kernel<<<tiles, WAVES_PER_BLOCK * 32, 0, stream>>>(x, W, b, out, M);
}